// FocalLoss_2052994367910
// MI455X (gfx1250) — compile-verified
//
#include <hip/hip_runtime.h>

#define BATCH 8
#define ANN 60
#define NCLS 8
#define TPB 256

typedef float v2f __attribute__((ext_vector_type(2)));
typedef float v8f __attribute__((ext_vector_type(8)));

// Exact f32 wave32 sum using V_WMMA_F32_16X16X4_F32.
// A[m][0] = q_m (lanes 0-15, vgpr0), A[m][2] = q_{m+16} (lanes 16-31, vgpr0),
// A[m][1] = A[m][3] = 0.  B = all ones (layout independent).
// => D[m][n] = q_m + q_{m+16} for every n.  Sum the 8 D vgprs, then combine
// the two half-wave partials with one xor-16 shuffle.  EXEC must be full.
__device__ __forceinline__ float wave_sum32(float q) {
    v2f A; A.x = q;    A.y = 0.0f;
    v2f B; B.x = 1.0f; B.y = 1.0f;
    v8f C = {0.f, 0.f, 0.f, 0.f, 0.f, 0.f, 0.f, 0.f};
    v8f D = __builtin_amdgcn_wmma_f32_16x16x4_f32(
        /*neg_a=*/false, A, /*neg_b=*/false, B,
        /*c_mod=*/(short)0, C, /*reuse_a=*/false, /*reuse_b=*/false);
    float s = D[0] + D[1] + D[2] + D[3] + D[4] + D[5] + D[6] + D[7];
    s += __shfl_xor(s, 16, 32);
    return s;   // full 32-lane sum in every lane
}

__device__ __forceinline__ float huber9(float x) {
    float d = fabsf(x);
    return (d <= (1.0f / 9.0f)) ? (4.5f * d * d) : (d - (1.0f / 18.0f));
}

__global__ void fl_init(float* accum) {
    if (threadIdx.x < BATCH * 4) accum[threadIdx.x] = 0.0f;
}

__global__ __launch_bounds__(TPB) void fl_main(
    const float* __restrict__ cls,     // (B, N, 8)
    const float* __restrict__ reg,     // (B, N, 12)
    const float* __restrict__ anch,    // (N, 4)
    const float* __restrict__ ann,     // (B, 60, ann_stride)
    float* __restrict__ accum,         // (B, 4): cls, reg, vp, pos_cnt
    int N, int ann_stride)
{
    __shared__ float s_bx0[ANN], s_by0[ANN], s_bx1[ANN], s_by1[ANN], s_area[ANN];
    __shared__ float s_t[ANN][20];
    __shared__ float s_tv[ANN][6];
    __shared__ int   s_cid[ANN];
    __shared__ float s_red[TPB / 32][4];

    const int tid = threadIdx.x;
    const int b   = blockIdx.y;

    // ---- per-block annotation prep (60 annotations -> LDS) ----
    if (tid < ANN) {
        const float* ap = ann + ((size_t)b * ANN + tid) * (size_t)ann_stride;
        float t[21];
#pragma unroll
        for (int k = 0; k < 21; k++) t[k] = ap[k];
#pragma unroll
        for (int k = 0; k < 20; k++) s_t[tid][k] = t[k];
        float xmn = t[0], xmx = t[0], ymn = t[1], ymx = t[1];
#pragma unroll
        for (int k = 1; k < 8; k++) {               // even idx 0..14: fx & bx
            xmn = fminf(xmn, t[2 * k]);     xmx = fmaxf(xmx, t[2 * k]);
            ymn = fminf(ymn, t[2 * k + 1]); ymx = fmaxf(ymx, t[2 * k + 1]);
        }
        s_bx0[tid] = xmn; s_by0[tid] = ymn; s_bx1[tid] = xmx; s_by1[tid] = ymx;
        s_area[tid] = (xmx - xmn) * (ymx - ymn);
        s_tv[tid][0] = (t[4] + t[6] + t[12] + t[14] - (t[0] + t[2] + t[8] + t[10])) * 0.25f;
        s_tv[tid][1] = (t[5] + t[7] + t[13] + t[15] - (t[1] + t[3] + t[9] + t[11])) * 0.25f;
        s_tv[tid][2] = (t[2] + t[6] + t[10] + t[14] - (t[0] + t[4] + t[8] + t[12])) * 0.25f;
        s_tv[tid][3] = (t[3] + t[7] + t[11] + t[15] - (t[1] + t[5] + t[9] + t[13])) * 0.25f;
        s_tv[tid][4] = (t[0] + t[2] + t[4] + t[6] - (t[8] + t[10] + t[12] + t[14])) * 0.25f;
        s_tv[tid][5] = (t[1] + t[3] + t[5] + t[7] - (t[9] + t[11] + t[13] + t[15])) * 0.25f;
        s_cid[tid] = (int)t[20];
    }
    __syncthreads();

    // ---- per-anchor work (branchless masking; inactive lanes contribute 0) ----
    const int  i      = blockIdx.x * TPB + tid;
    const bool active = (i < N);
    const int  ii     = active ? i : (N - 1);

    const float4 a4  = ((const float4*)anch)[ii];
    const float aw   = a4.z - a4.x, ah = a4.w - a4.y;
    const float acx  = a4.x + 0.5f * aw, acy = a4.y + 0.5f * ah;
    const float areaA = aw * ah;

    // IoU argmax without per-box division (cross-multiplied compare; ua > 0)
    float binter = -1.0f, bua = 1.0f;
    int bestj = 0;
#pragma unroll 4
    for (int j = 0; j < ANN; j++) {
        float iw = fminf(a4.z, s_bx1[j]) - fmaxf(a4.x, s_bx0[j]);
        float ih = fminf(a4.w, s_by1[j]) - fmaxf(a4.y, s_by0[j]);
        iw = fmaxf(iw, 0.0f); ih = fmaxf(ih, 0.0f);
        float inter = iw * ih;
        float ua = fmaxf(areaA + s_area[j] - inter, 1e-8f);
        bool better = inter * bua > binter * ua;     // strict > keeps first argmax
        if (better) { binter = inter; bua = ua; bestj = j; }
    }
    const float iou_max = binter / bua;

    const bool pos   = active && (iou_max >= 0.5f);
    const bool valid = active && ((iou_max < 0.4f) || pos);
    const int  cid   = s_cid[bestj];

    // ---- focal classification loss ----
    const float* cp = cls + ((size_t)b * N + ii) * NCLS;
    const float4 c01 = ((const float4*)cp)[0];
    const float4 c23 = ((const float4*)cp)[1];
    float cv[NCLS] = {c01.x, c01.y, c01.z, c01.w, c23.x, c23.y, c23.z, c23.w};
    float cls_sum = 0.0f;
#pragma unroll
    for (int c = 0; c < NCLS; c++) {
        float p = fminf(fmaxf(cv[c], 1e-4f), 1.0f - 1e-4f);
        bool is1   = pos && (c == cid);
        float alpha = is1 ? 0.25f : 0.75f;
        float pt    = is1 ? (1.0f - p) : p;
        float bce   = -logf(is1 ? p : (1.0f - p));
        cls_sum += alpha * pt * pt * bce;
    }
    cls_sum = valid ? cls_sum : 0.0f;

    // ---- regression + VP losses (wave-uniform skip when no positives) ----
    float reg_l = 0.0f, vp_l = 0.0f;
    if (__any(pos)) {
        const float4* rp4 = (const float4*)(reg + ((size_t)b * N + ii) * 12);
        float4 q0 = rp4[0], q1 = rp4[1], q2 = rp4[2];
        float r[12] = {q0.x, q0.y, q0.z, q0.w, q1.x, q1.y, q1.z, q1.w,
                       q2.x, q2.y, q2.z, q2.w};

        // vanishing-point cosine terms
        float vps = 0.0f;
#pragma unroll
        for (int v = 0; v < 3; v++) {
            float rx = r[2 + 2 * v], ry = r[3 + 2 * v];
            float tx = s_tv[bestj][2 * v], ty = s_tv[bestj][2 * v + 1];
            float rn = sqrtf(rx * rx + ry * ry);
            float tn = sqrtf(tx * tx + ty * ty);
            float denom = pos ? (rn * tn) : 1.0f;
            vps += 1.0f - (rx * tx + ry * ty) / denom;
        }
        vp_l = pos ? (vps / 3.0f) : 0.0f;

        // smooth-L1 on 20 normalized targets
        const float s2a[8] = {-1, -1, 1, 1, -1, -1, 1, 1};
        const float s4a[8] = {-1, 1, -1, 1, -1, 1, -1, 1};
        const float s6a[8] = {1, 1, 1, 1, -1, -1, -1, -1};
        const float inv_aw = 1.0f / aw, inv_ah = 1.0f / ah;
        float sl = 0.0f;
#pragma unroll
        for (int k = 0; k < 8; k++) {
            float px = r[0] + s2a[k] * r[2] + s4a[k] * r[4] + s6a[k] * r[6];
            float py = r[1] + s2a[k] * r[3] + s4a[k] * r[5] + s6a[k] * r[7];
            float tnx = (s_t[bestj][2 * k]     - acx) * inv_aw;
            float tny = (s_t[bestj][2 * k + 1] - acy) * inv_ah;
            sl += huber9(tnx - px) + huber9(tny - py);
        }
#pragma unroll
        for (int k = 8; k < 10; k++) {
            float tnx = (s_t[bestj][2 * k]     - acx) * inv_aw;
            float tny = (s_t[bestj][2 * k + 1] - acy) * inv_ah;
            sl += huber9(tnx - r[8 + 2 * (k - 8)]) + huber9(tny - r[9 + 2 * (k - 8)]);
        }
        reg_l = pos ? (sl / 20.0f) : 0.0f;
    }

    // ---- block reduction: WMMA wave sums -> LDS -> one atomic per quantity ----
    float csum = wave_sum32(cls_sum);
    float rsum = wave_sum32(reg_l);
    float vsum = wave_sum32(vp_l);
    float psum = wave_sum32(pos ? 1.0f : 0.0f);

    const int wave = tid >> 5, lane = tid & 31;
    if (lane == 0) {
        s_red[wave][0] = csum; s_red[wave][1] = rsum;
        s_red[wave][2] = vsum; s_red[wave][3] = psum;
    }
    __syncthreads();
    if (tid < 4) {
        float acc = 0.0f;
#pragma unroll
        for (int w = 0; w < TPB / 32; w++) acc += s_red[w][tid];
        atomicAdd(&accum[b * 4 + tid], acc);
    }
}

__global__ void fl_finalize(const float* __restrict__ accum, float* __restrict__ out) {
    if (threadIdx.x == 0) {
        float cm = 0.0f, rm = 0.0f, vm = 0.0f;
#pragma unroll
        for (int b = 0; b < BATCH; b++) {
            float np = fmaxf(accum[b * 4 + 3], 1.0f);
            cm += accum[b * 4 + 0] / np;
            rm += accum[b * 4 + 1] / np;
            vm += accum[b * 4 + 2] / np;
        }
        out[0] = cm / (float)BATCH;
        out[1] = rm / (float)BATCH;
        out[2] = vm / (float)BATCH;
    }
}

extern "C" void kernel_launch(void* const* d_in, const int* in_sizes, int n_in,
                              void* d_out, int out_size, void* d_ws, size_t ws_size,
                              hipStream_t stream) {
    const float* cls  = (const float*)d_in[0];   // (8, N, 8)  f32
    const float* reg  = (const float*)d_in[1];   // (8, N, 12) f32
    const float* anch = (const float*)d_in[2];   // (1, N, 4)  f32
    const float* ann  = (const float*)d_in[3];   // (8, 60, S) f32
    float* out   = (float*)d_out;
    float* accum = (float*)d_ws;                 // 32 floats

    const int N          = in_sizes[2] / 4;
    const int ann_stride = in_sizes[3] / (BATCH * ANN);

    fl_init<<<1, 32, 0, stream>>>(accum);
    dim3 grid((N + TPB - 1) / TPB, BATCH);
    fl_main<<<grid, TPB, 0, stream>>>(cls, reg, anch, ann, accum, N, ann_stride);
    fl_finalize<<<1, 32, 0, stream>>>(accum, out);
}